// DigitCaps_42597485642004
// MI455X (gfx1250) — compile-verified
//
#include <hip/hip_runtime.h>
#include <hip/hip_bf16.h>
#include <math.h>

// ---------------------------------------------------------------------------
// CapsNet dynamic routing, fused & recomputed. f32 WMMA (16x16x4) on gfx1250.
// 3 heavy passes (u_hat recomputed each pass from x,W which live in L2),
// 3 tiny squash kernels. No 335MB u_hat materialization.
// ---------------------------------------------------------------------------

typedef __attribute__((ext_vector_type(2))) float v2f;
typedef __attribute__((ext_vector_type(8))) float v8f;

constexpr int kB   = 128;   // batch
constexpr int kR   = 4096;  // routes
constexpr int kC   = 10;    // capsules
constexpr int kO   = 16;    // out channels (capsule vector dim)
constexpr int kI   = 8;     // in channels (WMMA K = 2 x 4)
constexpr int kMT  = 16;    // batch tile (WMMA M)
constexpr int kNMT = kB / kMT;            // 8 batch tiles
constexpr int kRT  = 128;   // routes per block
constexpr int kNRT = kR / kRT;            // 32 route chunks
constexpr int kWAVES = 4;
constexpr int kRPW = kRT / kWAVES;        // 32 routes per wave
constexpr int kCO  = kC * kO;             // 160

// PHASE 0: c_ij = 1/10 uniform (iteration 0), no b_ij involvement.
// PHASE 1: b_new = u_hat . v0  (old b==0); store b_new; c = softmax(b_new).
// PHASE 2: b_new = b_old + u_hat . v1;     no store;   c = softmax(b_new).
template <int PHASE>
__global__ __launch_bounds__(kWAVES * 32) void caps_pass(
    const float* __restrict__ x,     // [B,R,I]
    const float* __restrict__ Wt,    // [R,C,O,I]
    const float* __restrict__ vprev, // [B,C,O] (unused for PHASE 0)
    float* __restrict__ bij,         // [R,B,C] workspace
    float* __restrict__ spart)       // [NRT,B,C,O] partial s per route chunk
{
  __shared__ float lds_v[kMT * kCO];           // staged v tile (10 KB)
  __shared__ float lds_s[kWAVES][kMT * kCO];   // per-wave s partials (40 KB)

  const int mtile  = blockIdx.x & (kNMT - 1);
  const int rchunk = blockIdx.x / kNMT;
  const int wave   = threadIdx.x >> 5;
  const int lane   = threadIdx.x & 31;
  const int o      = lane & 15;   // N index (out channel) / A-matrix M index
  const int half   = lane >> 4;   // lane half: selects K pair for A/B, M+8 for C/D

  if (PHASE != 0) {
    for (int e = threadIdx.x; e < kMT * kCO; e += blockDim.x) {
      int bl = e / kCO;
      lds_v[e] = vprev[(size_t)(mtile * kMT + bl) * kCO + (e % kCO)];
    }
    __syncthreads();
  }

  const v8f zero8 = {0.f, 0.f, 0.f, 0.f, 0.f, 0.f, 0.f, 0.f};
  v8f s_acc[kC];
#pragma unroll
  for (int c = 0; c < kC; ++c) s_acc[c] = zero8;

  const int bA = mtile * kMT + o; // A-matrix row = batch (lanes 0-15 == 16-31)
  const int k0 = half * 2;        // K offset within a K=4 step (ISA VGPR layout)

  for (int ri = 0; ri < kRPW; ++ri) {
    const int r = rchunk * kRT + wave * kRPW + ri;

    // ---- A matrix: x[bA][r][k] ; VGPR0 holds K=k0, VGPR1 holds K=k0+1 ----
    const float* xp = x + ((size_t)bA * kR + r) * kI;
    v2f a0 = *(const v2f*)(xp + k0);       // K-step 0 (i = 0..3)
    v2f a1 = *(const v2f*)(xp + k0 + 4);   // K-step 1 (i = 4..7)

    // ---- u_hat tile: acc[c] = 16x16 (batch x out-channel) for capsule c ----
    v8f acc[kC];
#pragma unroll
    for (int c = 0; c < kC; ++c) acc[c] = zero8;
#pragma unroll
    for (int c = 0; c < kC; ++c) {
      const float* wp = Wt + (((size_t)r * kC + c) * kO + o) * kI;
      v2f b0 = *(const v2f*)(wp + k0);     // B: K rows striped across halves
      v2f b1 = *(const v2f*)(wp + k0 + 4);
      acc[c] = __builtin_amdgcn_wmma_f32_16x16x4_f32(false, a0, false, b0,
                                                     (short)0, acc[c], false, false);
      acc[c] = __builtin_amdgcn_wmma_f32_16x16x4_f32(false, a1, false, b1,
                                                     (short)0, acc[c], false, false);
    }

    if (PHASE == 0) {
      // uniform c_ij = 1/10: accumulate raw, scale once at the end
#pragma unroll
      for (int c = 0; c < kC; ++c)
#pragma unroll
        for (int j = 0; j < 8; ++j) s_acc[c][j] += acc[c][j];
    } else {
      // agreement a[b,c] = sum_o u_hat[b,c,o] * v[b,c,o]  (16-lane butterfly)
      float bnew[8][kC];
#pragma unroll
      for (int j = 0; j < 8; ++j) {
        const int bl = half * 8 + j; // C/D layout: M = j + 8*half
#pragma unroll
        for (int c = 0; c < kC; ++c) {
          float p = acc[c][j] * lds_v[(bl * kC + c) * kO + o];
          p += __shfl_xor(p, 1);
          p += __shfl_xor(p, 2);
          p += __shfl_xor(p, 4);
          p += __shfl_xor(p, 8);
          bnew[j][c] = p;
        }
      }

      const size_t bbase = ((size_t)r * kB + (mtile * kMT + half * 8)) * kC;
      if (PHASE == 2) { // b_new = b_old + a  (broadcast loads within half-wave)
#pragma unroll
        for (int j = 0; j < 8; ++j)
#pragma unroll
          for (int c = 0; c < kC; ++c) bnew[j][c] += bij[bbase + j * kC + c];
      }
      if (PHASE == 1) { // store b_new (one lane per half-wave, values replicated)
        if (o == 0) {
#pragma unroll
          for (int j = 0; j < 8; ++j)
#pragma unroll
            for (int c = 0; c < kC; ++c) bij[bbase + j * kC + c] = bnew[j][c];
        }
      }

      // softmax over capsules, then weighted s accumulation
#pragma unroll
      for (int j = 0; j < 8; ++j) {
        float mx = bnew[j][0];
#pragma unroll
        for (int c = 1; c < kC; ++c) mx = fmaxf(mx, bnew[j][c]);
        float sum = 0.f;
#pragma unroll
        for (int c = 0; c < kC; ++c) {
          bnew[j][c] = __expf(bnew[j][c] - mx);
          sum += bnew[j][c];
        }
        const float inv = 1.0f / sum;
#pragma unroll
        for (int c = 0; c < kC; ++c)
          s_acc[c][j] += acc[c][j] * (bnew[j][c] * inv);
      }
    }
  }

  // ---- cross-wave reduction of s partials via LDS ----
#pragma unroll
  for (int c = 0; c < kC; ++c)
#pragma unroll
    for (int j = 0; j < 8; ++j) {
      const int bl = half * 8 + j;
      lds_s[wave][(bl * kC + c) * kO + o] =
          s_acc[c][j] * (PHASE == 0 ? 0.1f : 1.0f);
    }
  __syncthreads();
  for (int e = threadIdx.x; e < kMT * kCO; e += blockDim.x) {
    float sum = lds_s[0][e] + lds_s[1][e] + lds_s[2][e] + lds_s[3][e];
    const int bl = e / kCO;
    spart[((size_t)rchunk * kB + (mtile * kMT + bl)) * kCO + (e % kCO)] = sum;
  }
}

// Reduce route-chunk partials, add bias, squash. One thread per (b,c,o);
// the 16 o's of a capsule vector sit in one 16-lane group for the butterfly.
__global__ __launch_bounds__(256) void caps_squash(
    const float* __restrict__ spart, // [NRT,B,C,O]
    const float* __restrict__ bias,  // [C,O]
    float* __restrict__ vout)        // [B,C,O]
{
  const int idx = blockIdx.x * 256 + threadIdx.x; // < B*C*O
  float s = bias[idx % kCO];
  for (int k = 0; k < kNRT; ++k) s += spart[(size_t)k * (kB * kCO) + idx];
  float p = s * s;
  p += __shfl_xor(p, 1);
  p += __shfl_xor(p, 2);
  p += __shfl_xor(p, 4);
  p += __shfl_xor(p, 8); // p = squared_norm over the 16-dim capsule vector
  vout[idx] = s * p / ((1.0f + p) * sqrtf(p));
}

extern "C" void kernel_launch(void* const* d_in, const int* in_sizes, int n_in,
                              void* d_out, int out_size, void* d_ws, size_t ws_size,
                              hipStream_t stream) {
  const float* x    = (const float*)d_in[0]; // [128,4096,8]
  const float* Wt   = (const float*)d_in[1]; // [4096,10,16,8]
  const float* bias = (const float*)d_in[2]; // [10,16]
  float* out = (float*)d_out;                // [128,10,16,1]

  float* spart = (float*)d_ws;                        // 32*128*160 f32 (2.6 MB)
  float* vbuf  = spart + (size_t)kNRT * kB * kCO;     // 128*160 f32
  float* bijw  = vbuf + (size_t)kB * kCO;             // 4096*128*10 f32 (21 MB)

  const dim3 gp(kNMT * kNRT), bp(kWAVES * 32);
  const dim3 gs((kB * kCO) / 256), bs(256);

  caps_pass<0><<<gp, bp, 0, stream>>>(x, Wt, nullptr, bijw, spart);
  caps_squash<<<gs, bs, 0, stream>>>(spart, bias, vbuf);   // v0
  caps_pass<1><<<gp, bp, 0, stream>>>(x, Wt, vbuf, bijw, spart);
  caps_squash<<<gs, bs, 0, stream>>>(spart, bias, vbuf);   // v1
  caps_pass<2><<<gp, bp, 0, stream>>>(x, Wt, vbuf, bijw, spart);
  caps_squash<<<gs, bs, 0, stream>>>(spart, bias, out);    // v2 -> output
}